// GeoGraphSAGE_56581899157895
// MI455X (gfx1250) — compile-verified
//
#include <hip/hip_runtime.h>
#include <hip/hip_bf16.h>

typedef float v2f __attribute__((ext_vector_type(2)));
typedef float v8f __attribute__((ext_vector_type(8)));

static constexpr int N_NODES = 100000;
static constexpr int N_EDGES = 1600000;
static constexpr int FEAT    = 128;       // IN_C == HID == 128
static constexpr int OUT_C   = 16;
static constexpr float BN_EPS = 1e-5f;

// ---------------------------------------------------------------------------
// zero a float region
// ---------------------------------------------------------------------------
__global__ void zero_kernel(float* __restrict__ p, size_t n) {
    size_t i = (size_t)blockIdx.x * blockDim.x + threadIdx.x;
    if (i < n) p[i] = 0.0f;
}

// ---------------------------------------------------------------------------
// degree count (float atomics; exact for counts << 2^24)
// ---------------------------------------------------------------------------
__global__ void count_kernel(const int* __restrict__ dst, float* __restrict__ deg,
                             int n_edges) {
    int e = blockIdx.x * blockDim.x + threadIdx.x;
    if (e < n_edges) unsafeAtomicAdd(deg + dst[e], 1.0f);
}

__global__ void inv_deg_kernel(float* __restrict__ deg, int n) {
    int i = blockIdx.x * blockDim.x + threadIdx.x;
    if (i < n) deg[i] = 1.0f / fmaxf(deg[i], 1.0f);
}

// ---------------------------------------------------------------------------
// edge scatter: agg[dst] += h[src]   (one wave32 per edge, float4 per lane)
// ---------------------------------------------------------------------------
__global__ void scatter_kernel(const float* __restrict__ h,
                               const int* __restrict__ src,
                               const int* __restrict__ dst,
                               float* __restrict__ agg, int n_edges) {
    int gid  = blockIdx.x * blockDim.x + threadIdx.x;
    int edge = gid >> 5;          // 32 lanes per edge
    int lane = gid & 31;          // lane handles 4 consecutive floats
    if (edge >= n_edges) return;
    int s = src[edge];            // wave-uniform -> scalar load
    int d = dst[edge];
    const float4 v = *(const float4*)(h + (size_t)s * FEAT + lane * 4);
    float* base = agg + (size_t)d * FEAT + lane * 4;
    unsafeAtomicAdd(base + 0, v.x);
    unsafeAtomicAdd(base + 1, v.y);
    unsafeAtomicAdd(base + 2, v.z);
    unsafeAtomicAdd(base + 3, v.w);
}

// ---------------------------------------------------------------------------
// fused SAGE layer: out = (inv*agg) @ Wn + b + h @ Ws  [+ BN(eval) + ReLU]
// block = 16-node row tile; one wave per 16-wide N tile; K=128 in steps of 4
// using v_wmma_f32_16x16x4_f32 (full fp32 precision).
// ---------------------------------------------------------------------------
template <int F_OUT, bool DO_BN>
__global__ __launch_bounds__(32 * (F_OUT / 16))
void sage_layer_kernel(const float* __restrict__ agg,
                       const float* __restrict__ hin,
                       const float* __restrict__ inv_deg,
                       const float* __restrict__ Wn,   // [128 x F_OUT] row-major
                       const float* __restrict__ Ws,   // [128 x F_OUT] row-major
                       const float* __restrict__ bias, // [F_OUT]
                       const float* __restrict__ g,
                       const float* __restrict__ be,
                       const float* __restrict__ rm,
                       const float* __restrict__ rv,
                       float* __restrict__ out) {
    constexpr int NW = F_OUT / 16;           // waves per block
    __shared__ float As[16][FEAT];           // mean-scaled neighbor agg rows
    __shared__ float Hs[16][FEAT];           // self rows

    const int tile = blockIdx.x;             // 16 nodes per tile (100000/16 exact)
    const int tid  = threadIdx.x;

    // cooperative stage of 16 x 128 x 2 floats into LDS, scaling agg by 1/deg
    for (int i = tid; i < 16 * FEAT; i += 32 * NW) {
        int r = i >> 7, c = i & (FEAT - 1);
        int node = tile * 16 + r;
        float s = inv_deg[node];
        As[r][c] = agg[(size_t)node * FEAT + c] * s;
        Hs[r][c] = hin[(size_t)node * FEAT + c];
    }
    __syncthreads();

    const int wave    = tid >> 5;
    const int lane    = tid & 31;
    const int colbase = wave * 16;
    const int col     = lane & 15;           // N index for B/C/D fragments
    const int mrowlo  = lane & 15;           // M index for A fragment
    const int khalf   = (lane >> 4) * 2;     // lanes 16..31 hold K+2, K+3

    v8f acc = {};                             // 16x16 f32 accumulator fragment
#pragma unroll 4
    for (int k0 = 0; k0 < FEAT; k0 += 4) {
        const int ka = k0 + khalf;
        // neighbor path: (inv*agg) @ Wn
        v2f a0; a0.x = As[mrowlo][ka];  a0.y = As[mrowlo][ka + 1];
        v2f b0; b0.x = Wn[(size_t)ka * F_OUT + colbase + col];
                b0.y = Wn[(size_t)(ka + 1) * F_OUT + colbase + col];
        acc = __builtin_amdgcn_wmma_f32_16x16x4_f32(false, a0, false, b0,
                                                    (short)0, acc, false, false);
        // self path: h @ Ws
        v2f a1; a1.x = Hs[mrowlo][ka];  a1.y = Hs[mrowlo][ka + 1];
        v2f b1; b1.x = Ws[(size_t)ka * F_OUT + colbase + col];
                b1.y = Ws[(size_t)(ka + 1) * F_OUT + colbase + col];
        acc = __builtin_amdgcn_wmma_f32_16x16x4_f32(false, a1, false, b1,
                                                    (short)0, acc, false, false);
    }

    // per-column epilogue constants (bias, BN affine)
    const float bcol = bias[colbase + col];
    float scale = 1.0f, shift = 0.0f;
    if (DO_BN) {
        const float inv_std = rsqrtf(rv[colbase + col] + BN_EPS);
        scale = g[colbase + col] * inv_std;
        shift = be[colbase + col] - rm[colbase + col] * scale;
    }

    // C/D layout: VGPR j -> M=j (lanes 0-15) or M=j+8 (lanes 16-31); lane -> N
    const int mhi = (lane >> 4) * 8;
#pragma unroll
    for (int j = 0; j < 8; ++j) {
        const int node = tile * 16 + j + mhi;
        float v = acc[j] + bcol;
        if (DO_BN) v = fmaxf(v * scale + shift, 0.0f);
        out[(size_t)node * F_OUT + colbase + col] = v;
    }
}

// ---------------------------------------------------------------------------
// launch
// ---------------------------------------------------------------------------
extern "C" void kernel_launch(void* const* d_in, const int* in_sizes, int n_in,
                              void* d_out, int out_size, void* d_ws, size_t ws_size,
                              hipStream_t stream) {
    const float* x   = (const float*)d_in[0];
    const int*   ei  = (const int*)d_in[1];
    const int*   src = ei;
    const int*   dst = ei + N_EDGES;
    const float* Wn0 = (const float*)d_in[2];
    const float* b0  = (const float*)d_in[3];
    const float* Ws0 = (const float*)d_in[4];
    const float* Wn1 = (const float*)d_in[5];
    const float* b1  = (const float*)d_in[6];
    const float* Ws1 = (const float*)d_in[7];
    const float* Wn2 = (const float*)d_in[8];
    const float* b2  = (const float*)d_in[9];
    const float* Ws2 = (const float*)d_in[10];
    const float* g0  = (const float*)d_in[11];
    const float* be0 = (const float*)d_in[12];
    const float* rm0 = (const float*)d_in[13];
    const float* rv0 = (const float*)d_in[14];
    const float* g1  = (const float*)d_in[15];
    const float* be1 = (const float*)d_in[16];
    const float* rm1 = (const float*)d_in[17];
    const float* rv1 = (const float*)d_in[18];

    // workspace layout (floats): inv_deg [padded N] | agg [N*128] | h [N*128]
    constexpr size_t N_PAD = 100352;                 // 100000 -> 1024 multiple
    float* inv_deg = (float*)d_ws;
    float* agg     = inv_deg + N_PAD;
    float* hbuf    = agg + (size_t)N_NODES * FEAT;

    const int TB = 256;
    const size_t zero_head = N_PAD + (size_t)N_NODES * FEAT;   // inv_deg + agg
    const size_t agg_elems = (size_t)N_NODES * FEAT;
    const int row_tiles    = N_NODES / 16;                     // 6250 exact
    const int scat_blocks  = (N_EDGES * 32 + TB - 1) / TB;
    const int edge_blocks  = (N_EDGES + TB - 1) / TB;

    // degree (shared by all layers) + first agg zero
    zero_kernel<<<(int)((zero_head + TB - 1) / TB), TB, 0, stream>>>(inv_deg, zero_head);
    count_kernel<<<edge_blocks, TB, 0, stream>>>(dst, inv_deg, N_EDGES);
    inv_deg_kernel<<<(N_NODES + TB - 1) / TB, TB, 0, stream>>>(inv_deg, N_NODES);

    // ---- layer 0: x -> hbuf (BN + ReLU) ----
    scatter_kernel<<<scat_blocks, TB, 0, stream>>>(x, src, dst, agg, N_EDGES);
    sage_layer_kernel<FEAT, true><<<row_tiles, 256, 0, stream>>>(
        agg, x, inv_deg, Wn0, Ws0, b0, g0, be0, rm0, rv0, hbuf);

    // ---- layer 1: hbuf -> hbuf in place (row-local, safe) ----
    zero_kernel<<<(int)((agg_elems + TB - 1) / TB), TB, 0, stream>>>(agg, agg_elems);
    scatter_kernel<<<scat_blocks, TB, 0, stream>>>(hbuf, src, dst, agg, N_EDGES);
    sage_layer_kernel<FEAT, true><<<row_tiles, 256, 0, stream>>>(
        agg, hbuf, inv_deg, Wn1, Ws1, b1, g1, be1, rm1, rv1, hbuf);

    // ---- layer 2: hbuf -> d_out (no BN/ReLU) ----
    zero_kernel<<<(int)((agg_elems + TB - 1) / TB), TB, 0, stream>>>(agg, agg_elems);
    scatter_kernel<<<scat_blocks, TB, 0, stream>>>(hbuf, src, dst, agg, N_EDGES);
    sage_layer_kernel<OUT_C, false><<<row_tiles, 32, 0, stream>>>(
        agg, hbuf, inv_deg, Wn2, Ws2, b2, g0, be0, rm0, rv0, (float*)d_out);
}